// mLSTMCell_86500641341530
// MI455X (gfx1250) — compile-verified
//
#include <hip/hip_runtime.h>
#include <hip/hip_bf16.h>
#include <math.h>

typedef __attribute__((ext_vector_type(2))) float v2f;
typedef __attribute__((ext_vector_type(8))) float v8f;

typedef __attribute__((address_space(1))) int glob_i32;   // global AS pointer elem
typedef __attribute__((address_space(3))) int lds_i32;    // LDS AS pointer elem

#define Bsz 512
#define Tsz 256
#define Isz 7
#define Hsz 64
#define CH  32      // time-chunk length (8 chunks over T=256)
#define TPB 256     // 8 wave32s per workgroup; one workgroup per batch element

#if defined(__HIP_DEVICE_COMPILE__) && \
    __has_builtin(__builtin_amdgcn_global_load_async_to_lds_b32) && \
    __has_builtin(__builtin_amdgcn_s_wait_asynccnt)
#define HAVE_ASYNC_LDS 1
#else
#define HAVE_ASYNC_LDS 0
#endif

__device__ __forceinline__ float fast_rcp(float x) {
#if __has_builtin(__builtin_amdgcn_rcpf)
    return __builtin_amdgcn_rcpf(x);
#else
    return 1.0f / x;
#endif
}

__device__ __forceinline__ float fast_tanh(float x) {
#if __has_builtin(__builtin_amdgcn_tanhf)
    return __builtin_amdgcn_tanhf(x);
#else
    return tanhf(x);
#endif
}

__global__ __launch_bounds__(TPB)
void mlstm_fused_kernel(const float* __restrict__ x,
                        const float* __restrict__ C_in,
                        const float* __restrict__ n_in,
                        const float* __restrict__ Wq, const float* __restrict__ bq,
                        const float* __restrict__ Wk, const float* __restrict__ bk,
                        const float* __restrict__ Wv, const float* __restrict__ bv,
                        const float* __restrict__ Wi, const float* __restrict__ bi,
                        const float* __restrict__ Wf, const float* __restrict__ bfp,
                        const float* __restrict__ Wo, const float* __restrict__ bo,
                        float* __restrict__ out)
{
    __shared__ float xs[CH][8];
    __shared__ float gs[6][CH][Hsz];   // 0:q 1:k(scaled) 2:v 3:i(exp) 4:f(sig) 5:o(sig)
    __shared__ float qsum[CH];
    __shared__ float hs[CH * Hsz];
    __shared__ float nfin[Hsz];

    const int b    = blockIdx.x;
    const int tid  = threadIdx.x;
    const int lane = tid & 31;
    const int wave = tid >> 5;

    const float* Wg[6] = {Wq, Wk, Wv, Wi, Wf, Wo};
    const float* bg[6] = {bq, bk, bv, bi, bfp, bo};

    // ---- Per-thread slice of matrix memory C: row r, 16 columns c0..c0+15.
    const int r  = tid >> 2;
    const int c0 = (tid & 3) << 4;
    float Cst[16];
    #pragma unroll
    for (int j = 0; j < 16; ++j)
        Cst[j] = C_in[(size_t)b * Hsz * Hsz + r * Hsz + c0 + j];
    // n is row-invariant (zero init + broadcast update): keep column r as a scalar
    float n_r = n_in[(size_t)b * Hsz * Hsz + r];

    // ---- Loop-invariant WMMA B-fragments (weights), reused every chunk.
    // N = 6 gates * 64 = 24 tiles of 16; wave w owns tiles 3w..3w+2; K 7->8 pad.
    v2f bfr0[3], bfr1[3];
    float biasv[3], mulc[3];
    #pragma unroll
    for (int it = 0; it < 3; ++it) {
        int nt   = wave * 3 + it;
        int g    = nt >> 2;
        int colg = ((nt & 3) << 4) + (lane & 15);
        const float* W = Wg[g];
        int k0 = (lane < 16) ? 0 : 2;
        v2f f0; f0.x = W[colg * Isz + k0];
                f0.y = W[colg * Isz + k0 + 1];
        int k1 = k0 + 4;
        v2f f1; f1.x = W[colg * Isz + k1];
                f1.y = (k1 + 1 < Isz) ? W[colg * Isz + k1 + 1] : 0.0f;  // K=7 pad
        bfr0[it] = f0;
        bfr1[it] = f1;
        biasv[it] = bg[g][colg];
        mulc[it]  = (g == 1) ? 0.125f : 1.0f;   // k gets 1/sqrt(H)
    }

    for (int chunk = 0; chunk < Tsz / CH; ++chunk) {
        const int t0g = chunk * CH;

        // ---- stage x chunk into LDS (zero-pad k=7); async-to-LDS when available
        for (int idx = tid; idx < CH * 8; idx += TPB) {
            int t = idx >> 3, k = idx & 7;
            if (k < Isz) {
#if HAVE_ASYNC_LDS
                const float* gp = &x[((size_t)b * Tsz + t0g + t) * Isz + k];
                float* lp = &xs[t][k];
                __builtin_amdgcn_global_load_async_to_lds_b32(
                    (glob_i32*)(uintptr_t)gp,
                    (lds_i32*)(uintptr_t)lp,
                    0, 0);
#else
                xs[t][k] = x[((size_t)b * Tsz + t0g + t) * Isz + k];
#endif
            } else {
                xs[t][k] = 0.0f;
            }
        }
#if HAVE_ASYNC_LDS
        __builtin_amdgcn_s_wait_asynccnt(0);
#endif
        __syncthreads();

        if (chunk + 1 < Tsz / CH && tid == 0)
            __builtin_prefetch(&x[((size_t)b * Tsz + t0g + CH) * Isz], 0, 0);

        // ---- gate GEMM: [CH x 8] * [8 x 384] via v_wmma_f32_16x16x4_f32
        #pragma unroll
        for (int m = 0; m < CH / 16; ++m) {
            int ar = (m << 4) + (lane & 15);
            int ak = (lane < 16) ? 0 : 2;
            v2f a0; a0.x = xs[ar][ak];     a0.y = xs[ar][ak + 1];
            v2f a1; a1.x = xs[ar][ak + 4]; a1.y = xs[ar][ak + 5];
            #pragma unroll
            for (int it = 0; it < 3; ++it) {
                int nt   = wave * 3 + it;
                int g    = nt >> 2;
                int colg = ((nt & 3) << 4) + (lane & 15);
                v8f acc = {};
                acc = __builtin_amdgcn_wmma_f32_16x16x4_f32(
                        false, a0, false, bfr0[it], (short)0, acc, false, false);
                acc = __builtin_amdgcn_wmma_f32_16x16x4_f32(
                        false, a1, false, bfr1[it], (short)0, acc, false, false);
                // gate id is wave-uniform: force into SGPR -> pure scalar branches
                int mode = __builtin_amdgcn_readfirstlane(g);
                int tlb  = (m << 4) + ((lane < 16) ? 0 : 8);
                if (mode == 3) {                       // i: exp
                    #pragma unroll
                    for (int j = 0; j < 8; ++j)
                        gs[g][tlb + j][colg] = __expf(acc[j] + biasv[it]);
                } else if (mode >= 4) {                // f,o: sigmoid
                    #pragma unroll
                    for (int j = 0; j < 8; ++j) {
                        float e = __expf(-(acc[j] + biasv[it]));
                        gs[g][tlb + j][colg] = fast_rcp(1.0f + e);
                    }
                } else {                               // q,k,v: linear (k scaled)
                    #pragma unroll
                    for (int j = 0; j < 8; ++j)
                        gs[g][tlb + j][colg] = (acc[j] + biasv[it]) * mulc[it];
                }
            }
        }
        __syncthreads();

        // ---- qsum[t] = sum_r q[t][r]; all 256 threads: 8 lanes per step
        {
            int step = tid >> 3, l8 = tid & 7;
            const float4* qp = reinterpret_cast<const float4*>(&gs[0][step][l8 * 8]);
            float4 a = qp[0], c = qp[1];
            float s = (a.x + a.y) + (a.z + a.w) + (c.x + c.y) + (c.z + c.w);
            s += __shfl_xor(s, 1, 32);
            s += __shfl_xor(s, 2, 32);
            s += __shfl_xor(s, 4, 32);
            if (l8 == 0) qsum[step] = s;
        }
        __syncthreads();

        // ---- sequential recurrence over the chunk; C stays in VGPRs
        for (int s = 0; s < CH; ++s) {
            const float4* qp = reinterpret_cast<const float4*>(&gs[0][s][c0]);
            const float4* kp = reinterpret_cast<const float4*>(&gs[1][s][c0]);
            const float4* ip = reinterpret_cast<const float4*>(&gs[3][s][c0]);
            const float4* fp = reinterpret_cast<const float4*>(&gs[4][s][c0]);
            float vr = gs[2][s][r];
            float partial = 0.0f;
            #pragma unroll
            for (int jj = 0; jj < 4; ++jj) {
                float4 qv = qp[jj], kv = kp[jj], iv = ip[jj], fv = fp[jj];
                int j = jj * 4;
                Cst[j+0] = fv.x * Cst[j+0] + (iv.x * kv.x) * vr;
                Cst[j+1] = fv.y * Cst[j+1] + (iv.y * kv.y) * vr;
                Cst[j+2] = fv.z * Cst[j+2] + (iv.z * kv.z) * vr;
                Cst[j+3] = fv.w * Cst[j+3] + (iv.w * kv.w) * vr;
                partial += Cst[j+0] * qv.x + Cst[j+1] * qv.y
                         + Cst[j+2] * qv.z + Cst[j+3] * qv.w;
            }
            partial += __shfl_xor(partial, 1, 32);           // quad reduce (wave32)
            partial += __shfl_xor(partial, 2, 32);
            n_r = gs[4][s][r] * n_r + gs[3][s][r] * gs[1][s][r];
            if ((tid & 3) == 0) {
                float den = fmaxf(n_r * qsum[s], 1.0f);
                hs[s * Hsz + r] = gs[5][s][r] * fast_tanh(partial * fast_rcp(den));
            }
        }
        __syncthreads();

        // ---- coalesced h chunk writeback (float4 -> global_store_b128)
        {
            const float4* hp = reinterpret_cast<const float4*>(hs);
            float4* op = reinterpret_cast<float4*>(&out[((size_t)b * Tsz + t0g) * Hsz]);
            for (int idx = tid; idx < CH * Hsz / 4; idx += TPB)
                op[idx] = hp[idx];
        }
        __syncthreads();
    }

    // ---- final states: out = [ h | C | n ]
    const size_t COff = (size_t)Bsz * Tsz * Hsz;
    const size_t nOff = COff + (size_t)Bsz * Hsz * Hsz;
    {
        float4* cp = reinterpret_cast<float4*>(&out[COff + (size_t)b * Hsz * Hsz + r * Hsz + c0]);
        #pragma unroll
        for (int jj = 0; jj < 4; ++jj) {
            float4 v; v.x = Cst[jj*4+0]; v.y = Cst[jj*4+1];
                      v.z = Cst[jj*4+2]; v.w = Cst[jj*4+3];
            cp[jj] = v;
        }
    }
    if ((tid & 3) == 0) nfin[r] = n_r;
    __syncthreads();
    for (int idx = tid; idx < Hsz * Hsz; idx += TPB)
        out[nOff + (size_t)b * Hsz * Hsz + idx] = nfin[idx & 63];  // broadcast rows
}

extern "C" void kernel_launch(void* const* d_in, const int* in_sizes, int n_in,
                              void* d_out, int out_size, void* d_ws, size_t ws_size,
                              hipStream_t stream) {
    (void)in_sizes; (void)n_in; (void)out_size; (void)d_ws; (void)ws_size;
    const float* x  = (const float*)d_in[0];
    const float* C  = (const float*)d_in[1];
    const float* n  = (const float*)d_in[2];
    const float* Wq = (const float*)d_in[3];  const float* bq = (const float*)d_in[4];
    const float* Wk = (const float*)d_in[5];  const float* bk = (const float*)d_in[6];
    const float* Wv = (const float*)d_in[7];  const float* bv = (const float*)d_in[8];
    const float* Wi = (const float*)d_in[9];  const float* bi = (const float*)d_in[10];
    const float* Wf = (const float*)d_in[11]; const float* bf = (const float*)d_in[12];
    const float* Wo = (const float*)d_in[13]; const float* bo = (const float*)d_in[14];
    float* out = (float*)d_out;
    hipLaunchKernelGGL(mlstm_fused_kernel, dim3(Bsz), dim3(TPB), 0, stream,
                       x, C, n, Wq, bq, Wk, bk, Wv, bv, Wi, bi, Wf, bf, Wo, bo, out);
}